// PIFSA_GNN_44186623541857
// MI455X (gfx1250) — compile-verified
//
#include <hip/hip_runtime.h>
#include <math.h>

typedef float v2f __attribute__((ext_vector_type(2)));
typedef float v8f __attribute__((ext_vector_type(8)));

#define BDIM 128   // 4 wave32
#define ROWS 32    // batch rows per block
#define DD   64    // embedding dim
#define KN   8     // neighbors

__device__ __forceinline__ float sigmoidf_(float x) {
    return 1.0f / (1.0f + __expf(-x));
}

__global__ __launch_bounds__(BDIM)
void pifsa_gnn_fused(const int* __restrict__ u,
                     const int* __restrict__ v,
                     const int* __restrict__ uai,
                     const int* __restrict__ adj_ent,
                     const int* __restrict__ adj_rel,
                     const float* __restrict__ usr_emb,
                     const float* __restrict__ ent_emb,
                     const float* __restrict__ rel_emb,
                     const float* __restrict__ conv_w,
                     const float* __restrict__ conv_b,
                     const float* __restrict__ agg_w,
                     const float* __restrict__ agg_b,
                     float* __restrict__ out)
{
    // padded to 65 floats/row -> bank-conflict-free on 64 banks
    __shared__ float relL[32][65];     // all 32 relation embeddings
    __shared__ float aggWL[64][65];    // agg_w, row = d_in (K), col = d_out (N)
    __shared__ float hL[ROWS][65];     // self + neigh_agg
    __shared__ float ueL[ROWS][65];    // user embeddings (conv output)
    __shared__ float aggBL[64];
    __shared__ float outAcc[ROWS];

    const int tid = threadIdx.x;
    const int blockBase = blockIdx.x * ROWS;

    // ---------- Phase 1: stage small operands into LDS ----------
    for (int i = tid; i < 32 * DD; i += BDIM) relL[i >> 6][i & 63] = rel_emb[i];
    for (int i = tid; i < DD * DD; i += BDIM) aggWL[i >> 6][i & 63] = agg_w[i];
    if (tid < DD)   aggBL[tid]  = agg_b[tid];
    if (tid < ROWS) outAcc[tid] = 0.0f;

    // ---------- Phase 2: user side: gather * mul * 3-tap conv over D ----------
    {
        const int r  = tid >> 2;        // row within block (4 lanes per row)
        const int cg = tid & 3;
        const int c0 = cg * 16;         // this lane's 16 output columns
        const int b  = blockBase + r;

        const float* usr = usr_emb + (long)u[b] * DD;
        float usrv[18];
        #pragma unroll
        for (int t = 0; t < 18; ++t) {
            int dd = c0 - 1 + t;
            usrv[t] = (dd >= 0 && dd < DD) ? usr[dd] : 0.0f;
        }
        float acc[16];
        #pragma unroll
        for (int i = 0; i < 16; ++i) acc[i] = 0.0f;

        for (int k = 0; k < KN; ++k) {
            const float* ent = ent_emb + (long)uai[b * KN + k] * DD;
            // only the middle column of the 3x3 kernel touches real data (W==1)
            const float w0 = conv_w[k * 9 + 1];
            const float w1 = conv_w[k * 9 + 4];
            const float w2 = conv_w[k * 9 + 7];
            float ev[18];
            #pragma unroll
            for (int t = 0; t < 18; ++t) {
                int dd = c0 - 1 + t;
                ev[t] = (dd >= 0 && dd < DD) ? ent[dd] : 0.0f;
            }
            #pragma unroll
            for (int i = 0; i < 16; ++i) {
                acc[i] += usrv[i]   * ev[i]   * w0
                        + usrv[i+1] * ev[i+1] * w1
                        + usrv[i+2] * ev[i+2] * w2;
            }
        }
        const float cb = conv_b[0];
        #pragma unroll
        for (int i = 0; i < 16; ++i) ueL[r][c0 + i] = acc[i] + cb;
    }
    __syncthreads();

    // ---------- Phase 3: attention softmax + neighbor aggregation -> h ----------
    {
        const int r  = tid >> 2;
        const int cg = tid & 3;
        const int c0 = cg * 16;
        const int b  = blockBase + r;
        const int vb = v[b];

        int eidx[KN], ridx[KN];
        #pragma unroll
        for (int k = 0; k < KN; ++k) {
            eidx[k] = adj_ent[(long)vb * KN + k];
            ridx[k] = adj_rel[(long)vb * KN + k];
            __builtin_prefetch(ent_emb + (long)eidx[k] * DD + c0, 0, 0);
        }

        // scores: dot(user_emb[b], rel_emb[ridx[k]]) ; partial over 16 cols,
        // butterfly-reduce across the 4 lanes of this row
        float s[KN];
        #pragma unroll
        for (int k = 0; k < KN; ++k) {
            float p = 0.0f;
            #pragma unroll
            for (int i = 0; i < 16; ++i)
                p += ueL[r][c0 + i] * relL[ridx[k]][c0 + i];
            p += __shfl_xor(p, 1, 4);
            p += __shfl_xor(p, 2, 4);
            s[k] = p;
        }
        float m = s[0];
        #pragma unroll
        for (int k = 1; k < KN; ++k) m = fmaxf(m, s[k]);
        float e[KN];
        float sum = 0.0f;
        #pragma unroll
        for (int k = 0; k < KN; ++k) { e[k] = __expf(s[k] - m); sum += e[k]; }
        const float inv = 1.0f / sum;

        const float* selfp = ent_emb + (long)vb * DD + c0;
        float acc[16];
        #pragma unroll
        for (int i = 0; i < 16; ++i) acc[i] = selfp[i];
        for (int k = 0; k < KN; ++k) {
            const float a = e[k] * inv;
            const float* ep = ent_emb + (long)eidx[k] * DD + c0;
            #pragma unroll
            for (int i = 0; i < 16; ++i) acc[i] += a * ep[i];
        }
        #pragma unroll
        for (int i = 0; i < 16; ++i) hL[r][c0 + i] = acc[i];
    }
    __syncthreads();

    // ---------- Phase 4: item = sigmoid(h @ agg_w + b) via fp32 WMMA, fused final dot ----------
    {
        const int wave = tid >> 5;
        const int lane = tid & 31;
        const int half = lane >> 4;     // 0: lanes 0-15, 1: lanes 16-31
        const int l16  = lane & 15;
        const int rowBase = (wave >> 1) * 16;   // waves 0,1 -> rows 0..15; waves 2,3 -> 16..31

        #pragma unroll
        for (int nt = 0; nt < 2; ++nt) {
            const int n = (wave & 1) * 2 + nt;  // N-tile 0..3
            const int colBase = n * 16;
            v8f c = {};
            #pragma unroll
            for (int kk = 0; kk < 16; ++kk) {   // K = 64 as 16 steps of K=4
                v2f a, bf;
                const int arow = rowBase + l16;
                const int ak = kk * 4 + half * 2;   // lanes 0-15: K=0,1 ; 16-31: K=2,3
                a.x  = hL[arow][ak];
                a.y  = hL[arow][ak + 1];
                bf.x = aggWL[ak][colBase + l16];
                bf.y = aggWL[ak + 1][colBase + l16];
                c = __builtin_amdgcn_wmma_f32_16x16x4_f32(
                        false, a, false, bf, (short)0, c, false, false);
            }
            // c[j] holds result for row (rowBase + j + 8*half), col (colBase + l16)
            #pragma unroll
            for (int j = 0; j < 8; ++j) {
                const int row = rowBase + j + half * 8;
                const int col = colBase + l16;
                const float item = sigmoidf_(c[j] + aggBL[col]);
                float p = ueL[row][col] * item;
                p += __shfl_xor(p, 1, 16);
                p += __shfl_xor(p, 2, 16);
                p += __shfl_xor(p, 4, 16);
                p += __shfl_xor(p, 8, 16);
                if (l16 == 0) atomicAdd(&outAcc[row], p);  // ds_add_f32
            }
        }
    }
    __syncthreads();

    // ---------- Phase 5: final sigmoid ----------
    if (tid < ROWS) {
        out[blockBase + tid] = sigmoidf_(outAcc[tid]);
    }
}

extern "C" void kernel_launch(void* const* d_in, const int* in_sizes, int n_in,
                              void* d_out, int out_size, void* d_ws, size_t ws_size,
                              hipStream_t stream) {
    const int*   u        = (const int*)  d_in[0];
    const int*   v        = (const int*)  d_in[1];
    const int*   uai      = (const int*)  d_in[2];
    const int*   adj_ent  = (const int*)  d_in[3];
    const int*   adj_rel  = (const int*)  d_in[4];
    const float* usr_emb  = (const float*)d_in[5];
    const float* ent_emb  = (const float*)d_in[6];
    const float* rel_emb  = (const float*)d_in[7];
    const float* conv_w   = (const float*)d_in[8];
    const float* conv_b   = (const float*)d_in[9];
    const float* agg_w    = (const float*)d_in[10];
    const float* agg_b    = (const float*)d_in[11];
    float*       out      = (float*)d_out;

    const int B = in_sizes[0];          // 16384
    const int grid = B / ROWS;          // 512 blocks of 32 rows

    pifsa_gnn_fused<<<grid, BDIM, 0, stream>>>(
        u, v, uai, adj_ent, adj_rel, usr_emb, ent_emb, rel_emb,
        conv_w, conv_b, agg_w, agg_b, out);
}